// YoloLoss_13993003450617
// MI455X (gfx1250) — compile-verified
//
#include <hip/hip_runtime.h>

#define SG 14.0f
#define NBLK 1024
#define NTHR 256
#define NWAVE (NTHR / 32)

typedef float v2f __attribute__((ext_vector_type(2)));
typedef float v8f __attribute__((ext_vector_type(8)));

// ---------------------------------------------------------------------------
// Exact 32-lane sum through the CDNA5 matrix pipe.
// A (16x4 f32): lanes 0-15 VGPR0 hold K=0 (rows m=0..15), lanes 16-31 VGPR0
// hold K=2 (rows m=0..15). Put v in A[0], 0 in A[1]; B = ones. Then
// D[m][n] = v_m + v_{m+16}. Per-lane sum of the 8 D VGPRs gives the half-wave
// sums; one xor-16 shuffle completes the full wave sum in every lane.
// ---------------------------------------------------------------------------
__device__ __forceinline__ float wave_sum32_wmma(float v) {
  v2f a; a[0] = v;    a[1] = 0.0f;
  v2f b; b[0] = 1.0f; b[1] = 1.0f;
  v8f c = {};
  v8f d = __builtin_amdgcn_wmma_f32_16x16x4_f32(
      /*neg_a=*/false, a, /*neg_b=*/false, b,
      /*c_mod=*/(short)0, c, /*reuse_a=*/false, /*reuse_b=*/false);
  float t = d[0] + d[1] + d[2] + d[3] + d[4] + d[5] + d[6] + d[7];
  t += __shfl_xor(t, 16, 32);
  return t;
}

// ---------------------------------------------------------------------------
// Pass 1: streaming loss evaluation, one cell per thread (grid-stride),
// deterministic block reduction, one 4-float partial per block.
// ---------------------------------------------------------------------------
__global__ void __launch_bounds__(NTHR)
yolo_partial(const float* __restrict__ pred,
             const float* __restrict__ tbox,
             const float* __restrict__ tcls,
             const unsigned char* __restrict__ hasobj,
             float* __restrict__ partials,
             long long n_cells) {
  float cls_l = 0.0f, noobj_l = 0.0f, reg_l = 0.0f, cont_l = 0.0f;

  const long long stride = (long long)gridDim.x * NTHR;
  for (long long i = (long long)blockIdx.x * NTHR + threadIdx.x; i < n_cells;
       i += stride) {
    const float* p  = pred + i * 30;
    const float* tc = tcls + i * 20;

    // Prefetch next grid-stride iteration (speculative; OOB is dropped).
    __builtin_prefetch(p + 30 * stride, 0, 1);
    __builtin_prefetch(tc + 20 * stride, 0, 1);

    // pred: 30 floats, 8B-aligned -> 15 x b64 loads
    float pv[30];
#pragma unroll
    for (int j = 0; j < 15; ++j) {
      float2 v = reinterpret_cast<const float2*>(p)[j];
      pv[2 * j] = v.x;
      pv[2 * j + 1] = v.y;
    }
    // target cls: 20 floats, 16B-aligned -> 5 x b128 loads
    float tcv[20];
#pragma unroll
    for (int j = 0; j < 5; ++j) {
      float4 v = reinterpret_cast<const float4*>(tc)[j];
      tcv[4 * j] = v.x; tcv[4 * j + 1] = v.y;
      tcv[4 * j + 2] = v.z; tcv[4 * j + 3] = v.w;
    }
    float4 tb = reinterpret_cast<const float4*>(tbox)[i];
    float f = hasobj[i] ? 1.0f : 0.0f;

    // ---- class loss: f * sum (pred_cls - target_cls)^2
    float cl = 0.0f;
#pragma unroll
    for (int cch = 0; cch < 20; ++cch) {
      float dd = pv[10 + cch] - tcv[cch];
      cl = fmaf(dd, dd, cl);
    }
    cls_l += f * cl;

    // ---- no-object loss: (1-f) * (conf0^2 + conf1^2)
    float conf0 = pv[4], conf1 = pv[9];
    noobj_l += (1.0f - f) * (conf0 * conf0 + conf1 * conf1);

    // ---- target xyxy
    float tx = tb.x, ty = tb.y, tw = tb.z, th = tb.w;
    float txs = tx / SG, tys = ty / SG;
    float tx1 = txs - 0.5f * tw, ty1 = tys - 0.5f * th;
    float tx2 = txs + 0.5f * tw, ty2 = tys + 0.5f * th;
    float ta = (tx2 - tx1) * (ty2 - ty1);

    // ---- IoU for both boxes
    float iou[2];
#pragma unroll
    for (int bidx = 0; bidx < 2; ++bidx) {
      float x = pv[5 * bidx + 0], y = pv[5 * bidx + 1];
      float w = pv[5 * bidx + 2], h = pv[5 * bidx + 3];
      float xs = x / SG, ys = y / SG;
      float px1 = xs - 0.5f * w, py1 = ys - 0.5f * h;
      float px2 = xs + 0.5f * w, py2 = ys + 0.5f * h;
      float ltx = fmaxf(px1, tx1), lty = fmaxf(py1, ty1);
      float rbx = fminf(px2, tx2), rby = fminf(py2, ty2);
      float wx = fmaxf(rbx - ltx, 0.0f), wy = fmaxf(rby - lty, 0.0f);
      float inter = wx * wy;
      float pa = (px2 - px1) * (py2 - py1);
      iou[bidx] = inter / (pa + ta - inter);
    }

    // jnp.argmax ties -> first occurrence
    int best = (iou[1] > iou[0]) ? 1 : 0;
    float miou = fmaxf(iou[0], iou[1]);

    float bx = pv[5 * best + 0], by = pv[5 * best + 1];
    float bw = pv[5 * best + 2], bh = pv[5 * best + 3];
    float bconf = pv[5 * best + 4];

    float dx = bx - tx, dy = by - ty;
    float d_xy = dx * dx + dy * dy;
    float dw = sqrtf(fmaxf(bw, 0.0f)) - sqrtf(fmaxf(tw, 0.0f));
    float dh = sqrtf(fmaxf(bh, 0.0f)) - sqrtf(fmaxf(th, 0.0f));
    float d_wh = dw * dw + dh * dh;

    reg_l += f * (d_xy + d_wh);
    float dc = bconf - miou;
    cont_l += f * dc * dc;
  }

  // ---- deterministic block reduction: wave32 shuffle tree, then LDS
  float vals[4] = {cls_l, noobj_l, reg_l, cont_l};
#pragma unroll
  for (int cidx = 0; cidx < 4; ++cidx) {
    float v = vals[cidx];
#pragma unroll
    for (int off = 16; off > 0; off >>= 1) v += __shfl_down(v, off, 32);
    vals[cidx] = v;
  }

  __shared__ float sm[NWAVE * 4];
  int lane = threadIdx.x & 31;
  int wid = threadIdx.x >> 5;
  if (lane == 0) {
#pragma unroll
    for (int cidx = 0; cidx < 4; ++cidx) sm[wid * 4 + cidx] = vals[cidx];
  }
  __syncthreads();
  if (threadIdx.x == 0) {
    float s[4] = {0.0f, 0.0f, 0.0f, 0.0f};
#pragma unroll
    for (int w = 0; w < NWAVE; ++w)
#pragma unroll
      for (int cidx = 0; cidx < 4; ++cidx) s[cidx] += sm[w * 4 + cidx];
#pragma unroll
    for (int cidx = 0; cidx < 4; ++cidx)
      partials[(long long)blockIdx.x * 4 + cidx] = s[cidx];
  }
}

// ---------------------------------------------------------------------------
// Pass 2: one wave32 reduces the NBLK partials; cross-lane sums go through
// V_WMMA_F32_16X16X4_F32 (EXEC is all-1s: 32 threads, no divergence since
// NBLK % 32 == 0).
// ---------------------------------------------------------------------------
__global__ void __launch_bounds__(32)
yolo_final(const float* __restrict__ partials, float* __restrict__ out,
           int nblocks, float inv_n) {
  float acc[4] = {0.0f, 0.0f, 0.0f, 0.0f};
  for (int i = threadIdx.x; i < nblocks; i += 32) {
#pragma unroll
    for (int cidx = 0; cidx < 4; ++cidx)
      acc[cidx] += partials[(long long)i * 4 + cidx];
  }

  float cls_s   = wave_sum32_wmma(acc[0]);
  float noobj_s = wave_sum32_wmma(acc[1]);
  float reg_s   = wave_sum32_wmma(acc[2]);
  float cont_s  = wave_sum32_wmma(acc[3]);

  if (threadIdx.x == 0) {
    float total = cls_s + 0.5f * noobj_s + 5.0f * reg_s + cont_s;
    out[0] = total * inv_n;
    out[1] = reg_s * inv_n;
    out[2] = cont_s * inv_n;
    out[3] = noobj_s * inv_n;
    out[4] = cls_s * inv_n;
  }
}

extern "C" void kernel_launch(void* const* d_in, const int* in_sizes, int n_in,
                              void* d_out, int out_size, void* d_ws,
                              size_t ws_size, hipStream_t stream) {
  const float* pred = (const float*)d_in[0];
  const float* tbox = (const float*)d_in[1];
  const float* tcls = (const float*)d_in[2];
  const unsigned char* hobj = (const unsigned char*)d_in[3];  // jnp bool = u8
  float* out = (float*)d_out;

  long long n_cells = (long long)in_sizes[3];  // N * S * S
  int N = (int)(n_cells / (14 * 14));
  float inv_n = 1.0f / (float)N;

  float* partials = (float*)d_ws;  // NBLK * 4 floats = 16 KB scratch

  yolo_partial<<<NBLK, NTHR, 0, stream>>>(pred, tbox, tcls, hobj, partials,
                                          n_cells);
  yolo_final<<<1, 32, 0, stream>>>(partials, out, NBLK, inv_n);
}